// ViT_46368466927966
// MI455X (gfx1250) — compile-verified
//
#include <hip/hip_runtime.h>

// ViT patchify: images[128,3,224,224] f32 -> patches[128,196,768] f32.
// Pure permutation, HBM-bandwidth-bound (~154 MB traffic, ~6.6 us floor @ 23.3 TB/s).
// CDNA5 async copy engine does all data movement (no VGPR round trip):
//   global_load_async_to_lds_b128  (permuting gather, th:NT since input is single-use)
//   s_wait_asynccnt 0
//   global_store_async_from_lds_b128 (fully contiguous 512B/wave stores)
//
// Grid structure: one output patch row = 768 floats = 192 x 16B chunks.
//   blockDim = 192 (one chunk per thread per row), ROWS_PER_BLOCK = 4.
//   Lane-dependent address math is shift/mask only (computed once);
//   row -> (n, gi, gj) decomposition is uniform -> SALU.

#define THREADS 192
#define ROWS_PER_BLOCK 4

__global__ __launch_bounds__(THREADS)
void patchify_async_kernel(const float* __restrict__ in,
                           float* __restrict__ out,
                           unsigned total_rows) {
    // 4 rows * 192 chunks * 16B = 12 KB
    __shared__ __align__(16) char smem[ROWS_PER_BLOCK * THREADS * 16];

    const unsigned tid  = threadIdx.x;
    const unsigned row0 = blockIdx.x * ROWS_PER_BLOCK;

    // Per-lane input offset inside one image's (c,h,w) volume, in bytes.
    // chunk-in-row ci = tid: within = ci&3 (pw quarter), cph = ci>>2 = c*16+ph.
    const unsigned within   = tid & 3u;
    const unsigned cph      = tid >> 2;
    const unsigned c        = cph >> 4;          // 0..2
    const unsigned ph       = cph & 15u;         // 0..15
    const unsigned lane_in  = (c * 50176u + ph * 224u + within * 4u) * 4u;
    const unsigned lane_lds = (unsigned)(unsigned long long)smem + tid * 16u;

    // ---------------- Phase 1: async permuting gather  global -> LDS --------
#pragma unroll
    for (int j = 0; j < ROWS_PER_BLOCK; ++j) {
        unsigned row = row0 + (unsigned)j;            // uniform
        if (row < total_rows) {
            unsigned n  = row / 196u;                 // uniform -> SALU
            unsigned p  = row - n * 196u;
            unsigned gi = p / 14u;
            unsigned gj = p - gi * 14u;
            // floats: n*3*224*224 + gi*16*224 + gj*16 ; -> bytes
            unsigned sbase   = (n * 150528u + gi * 3584u + gj * 16u) * 4u;
            unsigned in_off  = sbase + lane_in;
            unsigned lds_addr = lane_lds + (unsigned)j * (THREADS * 16u);
            asm volatile("global_load_async_to_lds_b128 %0, %1, %2 th:TH_LOAD_NT"
                         :
                         : "v"(lds_addr), "v"(in_off), "s"(in)
                         : "memory");
        }
    }

    // Slots are wave-private: only our own async loads must land. No barrier.
    asm volatile("s_wait_asynccnt 0x0" ::: "memory");

    // ---------------- Phase 2: async contiguous stream  LDS -> global -------
#pragma unroll
    for (int j = 0; j < ROWS_PER_BLOCK; ++j) {
        unsigned row = row0 + (unsigned)j;
        if (row < total_rows) {
            unsigned out_off  = row * 3072u + tid * 16u;   // bytes, contiguous/wave
            unsigned lds_addr = lane_lds + (unsigned)j * (THREADS * 16u);
            asm volatile("global_store_async_from_lds_b128 %0, %1, %2"
                         :
                         : "v"(out_off), "v"(lds_addr), "s"(out)
                         : "memory");
        }
    }

    asm volatile("s_wait_asynccnt 0x0" ::: "memory");
}

extern "C" void kernel_launch(void* const* d_in, const int* in_sizes, int n_in,
                              void* d_out, int out_size, void* d_ws, size_t ws_size,
                              hipStream_t stream) {
    (void)in_sizes; (void)n_in; (void)d_ws; (void)ws_size;
    const float* images = (const float*)d_in[0];
    float* out = (float*)d_out;

    // out_size = 128*196*768 floats -> 25088 rows of 768 floats.
    unsigned total_rows = (unsigned)(out_size / 768);
    unsigned blocks = (total_rows + ROWS_PER_BLOCK - 1) / ROWS_PER_BLOCK;  // 6272

    patchify_async_kernel<<<blocks, THREADS, 0, stream>>>(images, out, total_rows);
}